// StandardMamba_42958262894624
// MI455X (gfx1250) — compile-verified
//
#include <hip/hip_runtime.h>
#include <hip/hip_bf16.h>
#include <stdint.h>

// ---------------- model constants ----------------
#define D_MODEL   768
#define D_STATE   16
#define D_INNER   1536          // EXPAND * D_MODEL
#define D_CONV    4
#define DT_RANK   48            // D_MODEL / 16
#define BATCH     2
#define SEQ       2048
#define MROWS     (BATCH * SEQ) // 4096
#define DBL_W     (DT_RANK + 2 * D_STATE) // 80
#define LN_EPS    1e-5f

typedef float v2f __attribute__((ext_vector_type(2)));
typedef float v8f __attribute__((ext_vector_type(8)));

// Async copy: global -> LDS, 16B per lane, tracked by ASYNCcnt.
__device__ __forceinline__ void async_b128(uint32_t lds_addr, const float* gptr) {
    asm volatile("global_load_async_to_lds_b128 %0, %1, off"
                 :: "v"(lds_addr), "v"(gptr) : "memory");
}
__device__ __forceinline__ void wait_async0() {
    asm volatile("s_wait_asynccnt 0x0" ::: "memory");
}
// generic->LDS byte offset (aperture: low 32 bits of the address)
__device__ __forceinline__ uint32_t lds_off(const void* p) {
    return (uint32_t)(uintptr_t)p;
}

// ============================================================
// Tiled FP32 WMMA GEMM:  C[M,N] = A[M,K] @ W[N,K]^T (+ Res)
// 256 threads = 8 waves; block tile 64(M) x 32(N); K slab 32.
// Double-buffered LDS filled with GLOBAL_LOAD_ASYNC_TO_LDS_B128,
// compute = V_WMMA_F32_16X16X4_F32 (full fp32 precision path).
// ============================================================
__global__ __launch_bounds__(256)
void gemm_f32_wmma_tiled(const float* __restrict__ A,
                         const float* __restrict__ W,
                         const float* __restrict__ Res,   // nullable residual
                         float* __restrict__ C,
                         int M, int N, int K)
{
    __shared__ __align__(16) float As[2][64][36];
    __shared__ __align__(16) float Ws[2][32][36];

    const int tid  = threadIdx.x;
    const int wave = tid >> 5;
    const int lane = tid & 31;
    const int bm   = blockIdx.y * 64;
    const int bn   = blockIdx.x * 32;
    const int mw   = (wave & 3) * 16;
    const int nw   = (wave >> 2) * 16;

    const int rowk = lane & 15;          // M (resp. N) index within tile
    const int kb   = (lane >> 4) << 1;   // lanes 0-15 -> K 0,1 ; lanes 16-31 -> K 2,3

    // cooperative-copy coordinates: 256 threads x float4
    const int ra = tid >> 3;             // 0..31
    const int c4 = (tid & 7) << 2;       // 0,4,...,28

    auto issue_slab = [&](int buf, int k0) {
        async_b128(lds_off(&As[buf][ra][c4]),
                   &A[(size_t)(bm + ra) * K + k0 + c4]);
        async_b128(lds_off(&As[buf][ra + 32][c4]),
                   &A[(size_t)(bm + ra + 32) * K + k0 + c4]);
        async_b128(lds_off(&Ws[buf][ra][c4]),
                   &W[(size_t)(bn + ra) * K + k0 + c4]);
    };

    v8f acc = {};
    int buf = 0;

    issue_slab(0, 0);
    wait_async0();
    __syncthreads();

    for (int k0 = 0; k0 < K; k0 += 32) {
        // prefetch next slab asynchronously into the other buffer
        if (k0 + 32 < K) issue_slab(buf ^ 1, k0 + 32);

        // ---- 8 x V_WMMA_F32_16X16X4_F32 over the 32-wide slab ----
#pragma unroll
        for (int kk = 0; kk < 32; kk += 4) {
            v2f a = { As[buf][mw + rowk][kk + kb], As[buf][mw + rowk][kk + kb + 1] };
            v2f b = { Ws[buf][nw + rowk][kk + kb], Ws[buf][nw + rowk][kk + kb + 1] };
            acc = __builtin_amdgcn_wmma_f32_16x16x4_f32(
                false, a, false, b, (short)0, acc, false, false);
        }

        wait_async0();       // our async copies into buf^1 have landed
        __syncthreads();     // everyone done reading buf, buf^1 published
        buf ^= 1;
    }

    // ---- epilogue: D layout -> VGPR r holds row r (lanes 0-15) / 8+r (lanes 16-31)
    const int colc  = lane & 15;
    const int rhalf = (lane >> 4) << 3;
#pragma unroll
    for (int r = 0; r < 8; ++r) {
        int gr = bm + mw + rhalf + r;
        int gc = bn + nw + colc;
        float v = acc[r];
        if (Res) v += Res[(size_t)gr * N + gc];
        C[(size_t)gr * N + gc] = v;
    }
}

// ============================================================
// Small FP32 WMMA GEMM: one wave per 16x16 tile, direct global loads.
// C[M,N] = A[M,K](lda) @ W[N,K]^T(ldw); mode 1: softplus(v + bias[n])
// ============================================================
__global__ __launch_bounds__(32)
void gemm_f32_wmma_small(const float* __restrict__ A, int lda,
                         const float* __restrict__ W, int ldw,
                         const float* __restrict__ bias,
                         float* __restrict__ C, int ldc,
                         int K, int mode)
{
    const int lane = threadIdx.x;
    const int bm   = blockIdx.y * 16;
    const int bn   = blockIdx.x * 16;
    const int rowk = lane & 15;
    const int kb   = (lane >> 4) << 1;

    v8f acc = {};
    const float* ap = &A[(size_t)(bm + rowk) * lda + kb];
    const float* wp = &W[(size_t)(bn + rowk) * ldw + kb];

    for (int k0 = 0; k0 < K; k0 += 16) {
        __builtin_prefetch(ap + k0 + 16, 0, 1);
        __builtin_prefetch(wp + k0 + 16, 0, 1);
#pragma unroll
        for (int k = k0; k < k0 + 16; k += 4) {
            float2 av = *reinterpret_cast<const float2*>(ap + k);
            float2 wv = *reinterpret_cast<const float2*>(wp + k);
            v2f a = { av.x, av.y };
            v2f b = { wv.x, wv.y };
            acc = __builtin_amdgcn_wmma_f32_16x16x4_f32(
                false, a, false, b, (short)0, acc, false, false);
        }
    }

    const int colc  = lane & 15;
    const int rhalf = (lane >> 4) << 3;
#pragma unroll
    for (int r = 0; r < 8; ++r) {
        int gr = bm + rhalf + r;
        int gc = bn + colc;
        float v = acc[r];
        if (mode == 1) {
            v += bias[gc];
            v = (v > 20.f) ? v : __logf(1.f + __expf(v));   // softplus
        }
        C[(size_t)gr * ldc + gc] = v;
    }
}

// ============================================================
// Causal depthwise conv (width 4) + SiLU over xi half of xz.
// xz: [B, L, 2*D_INNER]; u out: [B, L, D_INNER]
// ============================================================
__global__ __launch_bounds__(256)
void conv_silu_kernel(const float* __restrict__ xz,
                      const float* __restrict__ cw,
                      const float* __restrict__ cb,
                      float* __restrict__ u, int L)
{
    int idx = blockIdx.x * 256 + threadIdx.x;
    if (idx >= BATCH * SEQ * D_INNER) return;
    int d = idx % D_INNER;
    int t = (idx / D_INNER) % L;
    int b = idx / (D_INNER * L);

    const float* w = &cw[(size_t)d * D_CONV];
    float acc = cb[d];
#pragma unroll
    for (int j = 0; j < D_CONV; ++j) {
        int ts = t - (D_CONV - 1) + j;
        if (ts >= 0)
            acc += w[j] * xz[((size_t)(b * L + ts)) * (2 * D_INNER) + d];
    }
    float sig = 1.f / (1.f + __expf(-acc));
    u[(size_t)idx] = acc * sig;   // silu
}

// ============================================================
// Selective scan: one thread per (b, d) channel, 16-state recurrence.
// B_t / C_t chunks double-buffered in LDS via async copies: the next
// 64-step chunk streams in (ASYNCcnt) while the serial recurrence runs.
// Output gated by SiLU(z), written in place over u.
// ============================================================
#define SCAN_NBLK (D_INNER / 256)   // 6
#define TCHUNK    64

__global__ __launch_bounds__(256)
void scan_kernel(float* __restrict__ u,            // in: conv act, out: gated y
                 const float* __restrict__ dt,
                 const float* __restrict__ dbl,     // [BL, 80]: dt_r|B|C
                 const float* __restrict__ xz,      // z half at offset D_INNER
                 const float* __restrict__ A_log,
                 const float* __restrict__ Dskip,
                 int L)
{
    const int tid = threadIdx.x;
    const int b   = blockIdx.x / SCAN_NBLK;
    const int d   = (blockIdx.x % SCAN_NBLK) * 256 + tid;

    __shared__ __align__(16) float bc[2][TCHUNK * 32];

    // chunk copy coords: 512 float4 slots, 2 per thread
    auto issue_chunk = [&](int buf, int t0) {
#pragma unroll
        for (int j = 0; j < 2; ++j) {
            int i  = tid + j * 256;       // 0..511
            int tl = i >> 3;              // 0..63
            int s4 = (i & 7) << 2;        // 0,4,...,28
            async_b128(lds_off(&bc[buf][tl * 32 + s4]),
                       &dbl[((size_t)(b * L + t0 + tl)) * DBL_W + DT_RANK + s4]);
        }
    };

    float Areg[D_STATE];
    float h[D_STATE];
#pragma unroll
    for (int s = 0; s < D_STATE; ++s) {
        Areg[s] = -__expf(A_log[(size_t)d * D_STATE + s]);
        h[s] = 0.f;
    }
    const float dsk = Dskip[d];

    int buf = 0;
    issue_chunk(0, 0);
    wait_async0();
    __syncthreads();

    for (int t0 = 0; t0 < L; t0 += TCHUNK) {
        if (t0 + TCHUNK < L) issue_chunk(buf ^ 1, t0 + TCHUNK);

        for (int tl = 0; tl < TCHUNK; ++tl) {
            size_t row = (size_t)(b * L + t0 + tl);
            float dtv = dt[row * D_INNER + d];
            float uv  = u[row * D_INNER + d];
            float du  = dtv * uv;
            const float* Bt = &bc[buf][tl * 32];
            const float* Ct = Bt + D_STATE;
            float y = 0.f;
#pragma unroll
            for (int s = 0; s < D_STATE; ++s) {
                float dA = __expf(dtv * Areg[s]);       // TRANS unit
                h[s] = dA * h[s] + du * Bt[s];
                y   += h[s] * Ct[s];
            }
            y += uv * dsk;
            float zv  = xz[row * (2 * D_INNER) + D_INNER + d];
            float sig = 1.f / (1.f + __expf(-zv));
            u[row * D_INNER + d] = y * (zv * sig);      // gate, in place
        }

        wait_async0();
        __syncthreads();
        buf ^= 1;
    }
}

// ============================================================
// LayerNorm over last dim (768): one block per (b,t) row.
// ============================================================
__global__ __launch_bounds__(256)
void layernorm_kernel(const float* __restrict__ x,
                      const float* __restrict__ nw,
                      const float* __restrict__ nb,
                      float* __restrict__ out)
{
    __shared__ float s1[256];
    __shared__ float s2[256];
    const int tid = threadIdx.x;
    const size_t row = (size_t)blockIdx.x * D_MODEL;

    float v0 = x[row + tid];
    float v1 = x[row + tid + 256];
    float v2 = x[row + tid + 512];
    s1[tid] = v0 + v1 + v2;
    s2[tid] = v0 * v0 + v1 * v1 + v2 * v2;
    __syncthreads();
    for (int off = 128; off > 0; off >>= 1) {
        if (tid < off) { s1[tid] += s1[tid + off]; s2[tid] += s2[tid + off]; }
        __syncthreads();
    }
    const float mu   = s1[0] * (1.f / D_MODEL);
    const float var  = s2[0] * (1.f / D_MODEL) - mu * mu;
    const float rstd = rsqrtf(var + LN_EPS);

    out[row + tid]       = (v0 - mu) * rstd * nw[tid]       + nb[tid];
    out[row + tid + 256] = (v1 - mu) * rstd * nw[tid + 256] + nb[tid + 256];
    out[row + tid + 512] = (v2 - mu) * rstd * nw[tid + 512] + nb[tid + 512];
}

// ============================================================
// host-side launch
// ============================================================
extern "C" void kernel_launch(void* const* d_in, const int* in_sizes, int n_in,
                              void* d_out, int out_size, void* d_ws, size_t ws_size,
                              hipStream_t stream)
{
    const float* x_in      = (const float*)d_in[0];
    const float* in_proj_w = (const float*)d_in[1];
    const float* conv_w    = (const float*)d_in[2];
    const float* conv_b    = (const float*)d_in[3];
    const float* x_proj_w  = (const float*)d_in[4];
    const float* dt_proj_w = (const float*)d_in[5];
    const float* dt_proj_b = (const float*)d_in[6];
    const float* A_log     = (const float*)d_in[7];
    const float* D_skip    = (const float*)d_in[8];
    const float* out_proj_w= (const float*)d_in[9];
    const float* norm_w    = (const float*)d_in[10];
    const float* norm_b    = (const float*)d_in[11];

    // workspace layout (floats)
    float* ws = (float*)d_ws;
    const size_t n_xz  = (size_t)MROWS * 2 * D_INNER;  // 12.58M
    const size_t n_u   = (size_t)MROWS * D_INNER;      // 6.29M
    const size_t n_dt  = n_u;
    const size_t n_dbl = (size_t)MROWS * DBL_W;        // 0.33M
    const size_t n_xb  = (size_t)MROWS * D_MODEL;      // 3.15M

    float* xz  = ws;                  // [MROWS, 3072]
    float* u   = xz  + n_xz;          // [MROWS, 1536]  (conv out, later gated y)
    float* dtb = u   + n_u;           // [MROWS, 1536]
    float* dbl = dtb + n_dt;          // [MROWS, 80]
    float* xb0 = dbl + n_dbl;         // [MROWS, 768]
    float* xb1 = xb0 + n_xb;          // [MROWS, 768]

    for (int l = 0; l < 2; ++l) {
        const float* xin  = (l == 0) ? x_in : xb0;
        float*       xout = (l == 0) ? xb0  : xb1;

        const float* Win  = in_proj_w  + (size_t)l * (2 * D_INNER) * D_MODEL;
        const float* Wcv  = conv_w     + (size_t)l * D_INNER * D_CONV;
        const float* Bcv  = conv_b     + (size_t)l * D_INNER;
        const float* Wxp  = x_proj_w   + (size_t)l * DBL_W * D_INNER;
        const float* Wdt  = dt_proj_w  + (size_t)l * D_INNER * DT_RANK;
        const float* Bdt  = dt_proj_b  + (size_t)l * D_INNER;
        const float* Al   = A_log      + (size_t)l * D_INNER * D_STATE;
        const float* Dl   = D_skip     + (size_t)l * D_INNER;
        const float* Wout = out_proj_w + (size_t)l * D_MODEL * D_INNER;

        // 1) xz = x @ in_proj_w^T   [4096 x 3072 x 768]
        {
            dim3 g((2 * D_INNER) / 32, MROWS / 64);
            gemm_f32_wmma_tiled<<<g, 256, 0, stream>>>(
                xin, Win, nullptr, xz, MROWS, 2 * D_INNER, D_MODEL);
        }
        // 2) u = silu(causal_conv(xi))
        {
            int n = BATCH * SEQ * D_INNER;
            conv_silu_kernel<<<(n + 255) / 256, 256, 0, stream>>>(
                xz, Wcv, Bcv, u, SEQ);
        }
        // 3) dbl = u @ x_proj_w^T   [4096 x 80 x 1536]
        {
            dim3 g(DBL_W / 16, MROWS / 16);
            gemm_f32_wmma_small<<<g, 32, 0, stream>>>(
                u, D_INNER, Wxp, D_INNER, nullptr, dbl, DBL_W, D_INNER, 0);
        }
        // 4) dt = softplus(dt_r @ dt_proj_w^T + b)   [4096 x 1536 x 48]
        {
            dim3 g(D_INNER / 16, MROWS / 16);
            gemm_f32_wmma_small<<<g, 32, 0, stream>>>(
                dbl, DBL_W, Wdt, DT_RANK, Bdt, dtb, D_INNER, DT_RANK, 1);
        }
        // 5) selective scan + D skip + SiLU(z) gate (in-place on u)
        {
            scan_kernel<<<BATCH * SCAN_NBLK, 256, 0, stream>>>(
                u, dtb, dbl, xz, Al, Dl, SEQ);
        }
        // 6) x_out = yg @ out_proj_w^T + x_in   [4096 x 768 x 1536]
        {
            dim3 g(D_MODEL / 32, MROWS / 64);
            gemm_f32_wmma_tiled<<<g, 256, 0, stream>>>(
                u, Wout, xin, xout, MROWS, D_MODEL, D_INNER);
        }
    }

    // final layernorm -> d_out
    layernorm_kernel<<<MROWS, 256, 0, stream>>>(xb1, norm_w, norm_b, (float*)d_out);
}